// PerParticleDeterminantAntiequivarianceLayer_57458072485930
// MI455X (gfx1250) — compile-verified
//
#include <hip/hip_runtime.h>
#include <math.h>

// Problem dimensions (fixed by the reference)
#define B_      4096
#define NELEC   32
#define D_      512
#define NSPINS  2
#define NION    4
#define N_      16            // per-spin electrons == orbital matrix size

typedef float v2f __attribute__((ext_vector_type(2)));
typedef float v8f __attribute__((ext_vector_type(8)));

#define WAVES    8            // waves per block; one (s,b) tile per wave
#define K0STEPS  (D_ / 4)     // 128 WMMA K-steps (fp32 WMMA has K=4)

// ---- dynamic LDS layout (in floats) ----
#define SZ_WPACK   (K0STEPS * 32 * 2)       // 8192 floats = 32 KB per matrix
#define OFF_WROW   0
#define OFF_WCOL   (OFF_WROW + SZ_WPACK)
#define OFF_WAVE0  (OFF_WCOL + SZ_WPACK)    // 16384
#define PITCH17    17                       // 16x16 tiles padded to 17 (bank-conflict free)
#define SZ_TILE    (16 * PITCH17)           // 272
#define LANE_STRIDE 257                     // per-lane 16x16 GE matrix + 1 pad dword
#define SZ_GE      (16 * LANE_STRIDE)       // 4112
#define PER_WAVE   (3 * SZ_TILE + SZ_GE + 16)   // lin_p, lin_j, env, GE scratch, det[16]
#define SMEM_FLOATS (OFF_WAVE0 + WAVES * PER_WAVE)

__global__ __launch_bounds__(32 * WAVES) void ppdet_antiequiv_kernel(
    const float* __restrict__ eq,     // (B, NELEC, D)
    const float* __restrict__ r_ei,   // (B, NELEC, NION, 3)
    const float* __restrict__ Wrow,   // (S, D, N)
    const float* __restrict__ Wcol,   // (S, D, N)
    const float* __restrict__ bias,   // (S, N)
    const float* __restrict__ envA,   // (S, NION, N, 3, 3)
    const float* __restrict__ envPi,  // (S, NION, N)
    float* __restrict__ out)          // (S, B, N, D)
{
    extern __shared__ float smem[];
    const int tid  = threadIdx.x;
    const int wave = tid >> 5;
    const int lane = tid & 31;
    const int s    = blockIdx.x & 1;                       // spin uniform per block
    const int b    = (blockIdx.x >> 1) * WAVES + wave;     // batch per wave

    // ---------------------------------------------------------------
    // Stage W_row / W_col into LDS pre-permuted into the WMMA
    // B-matrix (4x16 fp32) fragment order: lane ln holds rows
    // K = 2*(ln/16)+{0,1}, column n = ln%16 -> one ds_load_b64 / frag.
    // ---------------------------------------------------------------
    {
        const float* wr = Wrow + (size_t)s * D_ * N_;
        const float* wc = Wcol + (size_t)s * D_ * N_;
        for (int idx = tid; idx < K0STEPS * 32; idx += blockDim.x) {
            const int k0 = idx >> 5;
            const int ln = idx & 31;
            const int d0 = 4 * k0 + 2 * (ln >> 4);
            const int n  = ln & 15;
            smem[OFF_WROW + 2 * idx + 0] = wr[(size_t)d0 * N_ + n];
            smem[OFF_WROW + 2 * idx + 1] = wr[(size_t)(d0 + 1) * N_ + n];
            smem[OFF_WCOL + 2 * idx + 0] = wc[(size_t)d0 * N_ + n];
            smem[OFF_WCOL + 2 * idx + 1] = wc[(size_t)(d0 + 1) * N_ + n];
        }
    }
    __syncthreads();

    const int lm = lane & 15;     // column / M-row-within-half
    const int lh = lane >> 4;     // lane half
    const float* xrow = eq + ((size_t)b * NELEC + (size_t)s * N_) * D_;

    // ---------------------------------------------------------------
    // GEMM phase: lin_p = x @ W_row ; lin_j = x @ W_col, fp32 WMMA.
    // A fragment (16x4 fp32): lane holds x[m=lm][k0*4 + 2*lh + {0,1}].
    // ---------------------------------------------------------------
    v8f accR = {};
    v8f accC = {};
    const float* aptr = xrow + (size_t)lm * D_ + 2 * lh;
    const v2f* brp = (const v2f*)(smem + OFF_WROW) + lane;
    const v2f* bcp = (const v2f*)(smem + OFF_WCOL) + lane;
    #pragma unroll 4
    for (int k0 = 0; k0 < K0STEPS; ++k0) {
        v2f a  = *(const v2f*)(aptr + 4 * k0);
        v2f br = brp[32 * k0];
        v2f bc = bcp[32 * k0];
        accR = __builtin_amdgcn_wmma_f32_16x16x4_f32(false, a, false, br,
                                                     (short)0, accR, false, false);
        accC = __builtin_amdgcn_wmma_f32_16x16x4_f32(false, a, false, bc,
                                                     (short)0, accC, false, false);
    }

    float* sLp  = smem + OFF_WAVE0 + wave * PER_WAVE;  // lin_p [16][17]
    float* sLj  = sLp + SZ_TILE;                       // lin_j + bias [16][17]
    float* sEv  = sLj + SZ_TILE;                       // env [16][17]
    float* sGE  = sEv + SZ_TILE;                       // per-lane LU scratch
    float* sDet = sGE + SZ_GE;                         // det[16]

    // ---------------------------------------------------------------
    // Envelope: env[j][k] = sum_I pi[I,k] * exp(-||A[I,k] @ r[b,j,I]||)
    // Each lane has fixed k = lm (matches the C-matrix layout), so the
    // 4 ion 3x3 matrices are loaded once per lane.
    // ---------------------------------------------------------------
    float Ae[NION][9];
    float piw[NION];
    #pragma unroll
    for (int I = 0; I < NION; ++I) {
        const float* ap = envA + (((size_t)s * NION + I) * N_ + lm) * 9;
        #pragma unroll
        for (int u = 0; u < 9; ++u) Ae[I][u] = ap[u];
        piw[I] = envPi[((size_t)s * NION + I) * N_ + lm];
    }
    const float bk = bias[(size_t)s * N_ + lm];
    const float* rbase = r_ei + ((size_t)b * NELEC + (size_t)s * N_) * NION * 3;

    #pragma unroll
    for (int g = 0; g < 8; ++g) {
        const int row = g + 8 * lh;     // C-layout: VGPR g -> M = g (+8 for hi half)
        const float* rp = rbase + (size_t)row * NION * 3;
        float e = 0.0f;
        #pragma unroll
        for (int I = 0; I < NION; ++I) {
            const float rx = rp[I * 3 + 0], ry = rp[I * 3 + 1], rz = rp[I * 3 + 2];
            const float wx = Ae[I][0] * rx + Ae[I][1] * ry + Ae[I][2] * rz;
            const float wy = Ae[I][3] * rx + Ae[I][4] * ry + Ae[I][5] * rz;
            const float wz = Ae[I][6] * rx + Ae[I][7] * ry + Ae[I][8] * rz;
            const float nrm = sqrtf(wx * wx + wy * wy + wz * wz);
            e += piw[I] * __expf(-nrm);
        }
        sLp[row * PITCH17 + lm] = accR[g];
        sLj[row * PITCH17 + lm] = accC[g] + bk;
        sEv[row * PITCH17 + lm] = e;
    }
    __syncthreads();

    // ---------------------------------------------------------------
    // Determinants: lane p (0..15) builds M_p[j][k] in its private
    // 257-dword-strided LDS region (bank-conflict free) and runs a
    // serial LU with partial pivoting. No cross-lane traffic.
    // ---------------------------------------------------------------
    if (lane < 16) {
        float* Mm = sGE + lane * LANE_STRIDE;
        #pragma unroll
        for (int j = 0; j < 16; ++j) {
            #pragma unroll
            for (int k = 0; k < 16; ++k) {
                Mm[j * 16 + k] = (sLp[lane * PITCH17 + k] + sLj[j * PITCH17 + k])
                                 * sEv[j * PITCH17 + k];
            }
        }
        float det = 1.0f;
        for (int i = 0; i < 16; ++i) {
            // partial pivot
            int pr = i;
            float bv = fabsf(Mm[i * 16 + i]);
            for (int rr = i + 1; rr < 16; ++rr) {
                const float v = fabsf(Mm[rr * 16 + i]);
                if (v > bv) { bv = v; pr = rr; }
            }
            if (pr != i) {
                det = -det;
                for (int k = i; k < 16; ++k) {
                    const float t = Mm[i * 16 + k];
                    Mm[i * 16 + k]  = Mm[pr * 16 + k];
                    Mm[pr * 16 + k] = t;
                }
            }
            const float pv = Mm[i * 16 + i];
            det *= pv;
            const float inv = 1.0f / pv;
            for (int rr = i + 1; rr < 16; ++rr) {
                const float f = Mm[rr * 16 + i] * inv;
                for (int k = i + 1; k < 16; ++k)
                    Mm[rr * 16 + k] -= f * Mm[i * 16 + k];
            }
        }
        sDet[lane] = det;
    }
    __syncthreads();

    // ---------------------------------------------------------------
    // Output: out[s][b][p][:] = x[p][:] * det[p]  (float4 streaming)
    // ---------------------------------------------------------------
    float* orow = out + (((size_t)s * B_ + b) * N_) * D_;
    #pragma unroll
    for (int p = 0; p < N_; ++p) {
        const float dv = sDet[p];
        const float4* xv4 = (const float4*)(xrow + (size_t)p * D_);
        float4* ov4       = (float4*)(orow + (size_t)p * D_);
        #pragma unroll
        for (int dd = 0; dd < 4; ++dd) {
            float4 xv = xv4[dd * 32 + lane];
            float4 ov;
            ov.x = xv.x * dv; ov.y = xv.y * dv;
            ov.z = xv.z * dv; ov.w = xv.w * dv;
            ov4[dd * 32 + lane] = ov;
        }
    }
}

extern "C" void kernel_launch(void* const* d_in, const int* in_sizes, int n_in,
                              void* d_out, int out_size, void* d_ws, size_t ws_size,
                              hipStream_t stream) {
    (void)in_sizes; (void)n_in; (void)out_size; (void)d_ws; (void)ws_size;
    const float* eq    = (const float*)d_in[0];
    const float* r_ei  = (const float*)d_in[1];
    const float* Wrow  = (const float*)d_in[2];
    const float* Wcol  = (const float*)d_in[3];
    const float* bias  = (const float*)d_in[4];
    const float* envA  = (const float*)d_in[5];
    const float* envPi = (const float*)d_in[6];
    float* out = (float*)d_out;

    dim3 grid((B_ / WAVES) * NSPINS);   // one spin per block, WAVES batches per block
    dim3 block(32 * WAVES);
    const size_t smem = (size_t)SMEM_FLOATS * sizeof(float);
    hipLaunchKernelGGL(ppdet_antiequiv_kernel, grid, block, smem, stream,
                       eq, r_ei, Wrow, Wcol, bias, envA, envPi, out);
}